// GCNModel_24223615549681
// MI455X (gfx1250) — compile-verified
//
#include <hip/hip_runtime.h>

// ---------------------------------------------------------------------------
// GCN (3 layers) for MI455X / gfx1250.
//   N=50000 nodes, E=800000 edges, D_IN=D_HID=96, D_OUT=64.
// Dense transforms use V_WMMA_F32_16X16X4_F32 (full fp32 matrix core path).
// Edge aggregation uses hardware global_atomic_add_f32; hW and the
// accumulator (19.2 MB each) are L2-resident (192 MB L2), so the scatter
// phase runs at L2 rate rather than HBM rate.
// ---------------------------------------------------------------------------

#define NN 50000
#define NE 800000
#define DIN 96
#define DHID 96
#define DOUT 64

typedef float v2f __attribute__((ext_vector_type(2)));
typedef float v8f __attribute__((ext_vector_type(8)));

__device__ __forceinline__ void atomicAddF(float* p, float v) {
    // Hardware FP32 atomic (global_atomic_add_f32), no CAS loop.
    unsafeAtomicAdd(p, v);
}

// ---------------------------------------------------------------- utilities
__global__ void k_zero(float* __restrict__ p, int n) {
    int i = blockIdx.x * blockDim.x + threadIdx.x;
    if (i < n) p[i] = 0.0f;
}

__global__ void k_relu(float* __restrict__ p, int n) {
    int i = blockIdx.x * blockDim.x + threadIdx.x;
    if (i < n) p[i] = fmaxf(p[i], 0.0f);
}

// ------------------------------------------------------------- edge norms
__global__ void k_degrees(const int* __restrict__ src, const int* __restrict__ tgt,
                          const float* __restrict__ w,
                          float* __restrict__ in_deg, float* __restrict__ out_deg) {
    int e = blockIdx.x * blockDim.x + threadIdx.x;
    if (e >= NE) return;
    float we = w[e];
    atomicAddF(&in_deg[src[e]], we);
    atomicAddF(&out_deg[tgt[e]], we);
}

__global__ void k_edgenorm(const int* __restrict__ src, const int* __restrict__ tgt,
                           const float* __restrict__ w,
                           const float* __restrict__ in_deg, const float* __restrict__ out_deg,
                           float* __restrict__ nw) {
    int e = blockIdx.x * blockDim.x + threadIdx.x;
    if (e >= NE) return;
    float di = in_deg[src[e]];
    float dq = out_deg[tgt[e]];
    float a = di > 0.0f ? rsqrtf(di) : 0.0f;
    float b = dq > 0.0f ? rsqrtf(dq) : 0.0f;
    nw[e] = w[e] * a * b;
}

// ---------------------------------------------------------------- fp32 WMMA GEMM
// Out[M x N] = A[M x 96] @ W[96 x N] + bias, N = NT*16, M = 50000 (3125 tiles).
// One wave (32 lanes) per 16-row M tile; A fragment reused across all NT
// column tiles. Layouts per CDNA5 ISA 7.12.2:
//   A (16x4 f32):  lane -> M = lane&15, kq = 2*(lane>>4); vgpr i -> K = kq+i
//   B (4x16 f32):  lane -> N = lane&15, same K striping as A
//   C/D (16x16):   vgpr r -> M = r + 8*(lane>>4), N = lane&15
template <int NT>
__launch_bounds__(32)
__global__ void k_gemm_wmma(const float* __restrict__ A,
                            const float* __restrict__ W,
                            const float* __restrict__ bias,
                            float* __restrict__ Out) {
    constexpr int K = 96;
    constexpr int N = NT * 16;

    const int lane = threadIdx.x & 31;
    const int half = lane >> 4;        // 0 or 1
    const int l15  = lane & 15;
    const int kq   = half * 2;         // K offset within 4-wide slab
    const size_t row0 = (size_t)blockIdx.x * 16;

    v8f acc[NT];
#pragma unroll
    for (int t = 0; t < NT; ++t) {
        float bv = bias[t * 16 + l15];
#pragma unroll
        for (int r = 0; r < 8; ++r) acc[t][r] = bv;
    }

    const float* Arow = A + (row0 + (size_t)l15) * K;
#pragma unroll
    for (int k = 0; k < K; k += 4) {
        v2f a;
        a.x = Arow[k + kq];
        a.y = Arow[k + kq + 1];
#pragma unroll
        for (int t = 0; t < NT; ++t) {
            v2f b;
            b.x = W[(size_t)(k + kq)     * N + t * 16 + l15];
            b.y = W[(size_t)(k + kq + 1) * N + t * 16 + l15];
            acc[t] = __builtin_amdgcn_wmma_f32_16x16x4_f32(
                false, a, false, b, (short)0, acc[t], false, false);
        }
    }

    const size_t mb = row0 + (size_t)(half * 8);
#pragma unroll
    for (int t = 0; t < NT; ++t)
#pragma unroll
        for (int r = 0; r < 8; ++r)
            Out[(mb + r) * N + t * 16 + l15] = acc[t][r];
}

// ------------------------------------------------------------ edge scatter
// out[src[e]][:] += hW[tgt[e]][:] * nw[e]; one thread per (edge, 4 features).
template <int D4>  // D/4 feature chunks (24 for D=96, 16 for D=64)
__global__ void k_aggregate(const float* __restrict__ hW,
                            const int* __restrict__ src, const int* __restrict__ tgt,
                            const float* __restrict__ nw,
                            float* __restrict__ out) {
    constexpr int D = D4 * 4;
    int tid = blockIdx.x * blockDim.x + threadIdx.x;
    if (tid >= NE * D4) return;
    int e = tid / D4;               // constant divisor -> mul/shift
    int c = (tid - e * D4) * 4;
    int s = src[e];
    int t = tgt[e];
    float w = nw[e];
    const float4 v = *reinterpret_cast<const float4*>(hW + (size_t)t * D + c);
    float* o = out + (size_t)s * D + c;
    atomicAddF(o + 0, v.x * w);
    atomicAddF(o + 1, v.y * w);
    atomicAddF(o + 2, v.z * w);
    atomicAddF(o + 3, v.w * w);
}

// ---------------------------------------------------------------------------
extern "C" void kernel_launch(void* const* d_in, const int* in_sizes, int n_in,
                              void* d_out, int out_size, void* d_ws, size_t ws_size,
                              hipStream_t stream) {
    const float* x    = (const float*)d_in[0];
    const int*   esrc = (const int*)  d_in[1];
    const int*   etgt = (const int*)  d_in[2];
    const float* ew   = (const float*)d_in[3];
    const float* W1   = (const float*)d_in[4];
    const float* b1   = (const float*)d_in[5];
    const float* W2   = (const float*)d_in[6];
    const float* b2   = (const float*)d_in[7];
    const float* W3   = (const float*)d_in[8];
    const float* b3   = (const float*)d_in[9];

    float* out  = (float*)d_out;
    float* ws   = (float*)d_ws;

    // workspace layout (floats)
    float* in_deg  = ws;                       // 50000
    float* out_deg = ws + NN;                  // 50000
    float* nw      = ws + 2 * NN;              // 800000
    float* bufA    = nw + NE;                  // 50000*96 (hW scratch)
    float* bufB    = bufA + (size_t)NN * DHID; // 50000*96 (layer-1 output)

    float* emb  = out;                         // 50000*96 node_embeddings
    float* pred = out + (size_t)NN * DHID;     // 50000*64 predictions

    const int T = 256;
    auto blk = [](long n, int t) { return (int)((n + t - 1) / t); };

    // ---- edge normalization ------------------------------------------------
    k_zero<<<blk(2 * NN, T), T, 0, stream>>>(in_deg, 2 * NN);
    k_degrees<<<blk(NE, T), T, 0, stream>>>(esrc, etgt, ew, in_deg, out_deg);
    k_edgenorm<<<blk(NE, T), T, 0, stream>>>(esrc, etgt, ew, in_deg, out_deg, nw);

    // ---- layer 1: x -> bufB ------------------------------------------------
    k_gemm_wmma<6><<<NN / 16, 32, 0, stream>>>(x, W1, b1, bufA);
    k_zero<<<blk((long)NN * DHID, T), T, 0, stream>>>(bufB, NN * DHID);
    k_aggregate<24><<<blk((long)NE * 24, T), T, 0, stream>>>(bufA, esrc, etgt, nw, bufB);
    k_relu<<<blk((long)NN * DHID, T), T, 0, stream>>>(bufB, NN * DHID);

    // ---- layer 2: bufB -> emb (d_out) --------------------------------------
    k_gemm_wmma<6><<<NN / 16, 32, 0, stream>>>(bufB, W2, b2, bufA);
    k_zero<<<blk((long)NN * DHID, T), T, 0, stream>>>(emb, NN * DHID);
    k_aggregate<24><<<blk((long)NE * 24, T), T, 0, stream>>>(bufA, esrc, etgt, nw, emb);
    k_relu<<<blk((long)NN * DHID, T), T, 0, stream>>>(emb, NN * DHID);

    // ---- layer 3: emb -> pred (d_out) --------------------------------------
    k_gemm_wmma<4><<<NN / 16, 32, 0, stream>>>(emb, W3, b3, bufA);
    k_zero<<<blk((long)NN * DOUT, T), T, 0, stream>>>(pred, NN * DOUT);
    k_aggregate<16><<<blk((long)NE * 16, T), T, 0, stream>>>(bufA, esrc, etgt, nw, pred);
    k_relu<<<blk((long)NN * DOUT, T), T, 0, stream>>>(pred, NN * DOUT);
}